// MyModel_52140902974179
// MI455X (gfx1250) — compile-verified
//
#include <hip/hip_runtime.h>
#include <math.h>

typedef __attribute__((ext_vector_type(2))) float v2f;
typedef __attribute__((ext_vector_type(8))) float v8f;

// D = A(16x4 f32) x B(4x16 f32) + C(16x16 f32), full fp32 WMMA on gfx1250.
static __device__ inline v8f wmma_f32(v2f a, v2f b, v8f c) {
  return __builtin_amdgcn_wmma_f32_16x16x4_f32(
      /*neg_a=*/false, a, /*neg_b=*/false, b,
      /*c_mod=*/(short)0, c, /*reuse_a=*/false, /*reuse_b=*/false);
}

// ---------------------------------------------------------------------------
// Build padded rhs in SPD basis: Y[i,b] = X[b, idx[i]] / sqrt(s[idx[i]]), 0 pad
// Y layout: [Np][Bp] row-major.
// ---------------------------------------------------------------------------
__global__ void build_rhs_kernel(const float* __restrict__ X,
                                 const float* __restrict__ s,
                                 const int* __restrict__ idx,
                                 float* __restrict__ Y,
                                 int V, int NI, int Bn, int Bp, int Np) {
  int t = blockIdx.x * blockDim.x + threadIdx.x;
  if (t >= Np * Bp) return;
  int row = t / Bp;
  int col = t - row * Bp;
  float v = 0.0f;
  if (row < NI && col < Bn) {
    int vx = idx[row];
    v = X[(size_t)col * V + vx] * (1.0f / sqrtf(s[vx]));
  }
  Y[t] = v;
}

// ---------------------------------------------------------------------------
// A = I + S^{1/2} (d0_i^T diag(w) d0_i) S^{1/2}, padded to Np with identity.
// Block tile 64x64, 128 threads = 4 waves, each wave owns a 32x32 quadrant
// (2x2 grid of 16x16 WMMA tiles). K streamed from d0 in chunks of 32 via LDS.
// ---------------------------------------------------------------------------
__global__ __launch_bounds__(128) void gemm_spd_kernel(
    const float* __restrict__ d0, const float* __restrict__ w,
    const float* __restrict__ s, const int* __restrict__ idx,
    float* __restrict__ A, int E, int V, int NI, int Np) {
  const int KC = 32, PIT = 65;  // [KC][64] panels, padded pitch
  extern __shared__ float smem[];
  float* sa = smem;             // sa[k*PIT + c] = d0[e0+k, idx[m0+c]]
  float* sb = smem + KC * PIT;  // sb[k*PIT + c] = w[e0+k]*d0[e0+k, idx[n0+c]]

  const int tid = threadIdx.x;
  const int lane = tid & 31;
  const int wave = tid >> 5;
  const int waveM = wave >> 1;  // 0..1
  const int waveN = wave & 1;   // 0..1
  const int lhalf = lane >> 4;  // selects K pair (0 -> K0,K1 ; 1 -> K2,K3)
  const int l16 = lane & 15;
  const int m0 = blockIdx.y * 64;
  const int n0 = blockIdx.x * 64;

  const v8f vzero = {0.f, 0.f, 0.f, 0.f, 0.f, 0.f, 0.f, 0.f};
  v8f acc[2][2];
#pragma unroll
  for (int i = 0; i < 2; ++i)
#pragma unroll
    for (int j = 0; j < 2; ++j) acc[i][j] = vzero;

  for (int e0 = 0; e0 < E; e0 += KC) {
#pragma unroll
    for (int i = 0; i < (KC * 64) / 128; ++i) {
      int flat = tid + i * 128;
      int k = flat >> 6;
      int c = flat & 63;
      int e = e0 + k;
      float va = 0.0f, vb = 0.0f;
      if (e < E) {
        int gm = m0 + c;
        int gn = n0 + c;
        if (gm < NI) va = d0[(size_t)e * V + idx[gm]];
        if (gn < NI) vb = w[e] * d0[(size_t)e * V + idx[gn]];
      }
      sa[k * PIT + c] = va;
      sb[k * PIT + c] = vb;
    }
    __syncthreads();
#pragma unroll
    for (int ks = 0; ks < KC; ks += 4) {
      int kb = ks + lhalf * 2;
      v2f av[2], bv[2];
#pragma unroll
      for (int tm = 0; tm < 2; ++tm) {
        int mm = waveM * 32 + tm * 16 + l16;
        av[tm].x = sa[kb * PIT + mm];
        av[tm].y = sa[(kb + 1) * PIT + mm];
      }
#pragma unroll
      for (int tn = 0; tn < 2; ++tn) {
        int nn = waveN * 32 + tn * 16 + l16;
        bv[tn].x = sb[kb * PIT + nn];
        bv[tn].y = sb[(kb + 1) * PIT + nn];
      }
#pragma unroll
      for (int tm = 0; tm < 2; ++tm)
#pragma unroll
        for (int tn = 0; tn < 2; ++tn)
          acc[tm][tn] = wmma_f32(av[tm], bv[tn], acc[tm][tn]);
    }
    __syncthreads();
  }

#pragma unroll
  for (int tm = 0; tm < 2; ++tm) {
#pragma unroll
    for (int tn = 0; tn < 2; ++tn) {
      int nn = n0 + waveN * 32 + tn * 16 + l16;
      float sn = (nn < NI) ? sqrtf(s[idx[nn]]) : 0.0f;
#pragma unroll
      for (int r = 0; r < 8; ++r) {
        int mm = m0 + waveM * 32 + tm * 16 + r + 8 * lhalf;
        float v = (mm == nn) ? 1.0f : 0.0f;
        if (mm < NI && nn < NI)
          v += acc[tm][tn][r] * sqrtf(s[idx[mm]]) * sn;
        A[(size_t)mm * Np + nn] = v;
      }
    }
  }
}

// ---------------------------------------------------------------------------
// Unblocked Cholesky of the 128x128 diagonal block (one workgroup, LDS-resident)
// ---------------------------------------------------------------------------
__global__ __launch_bounds__(128) void potrf_kernel(float* __restrict__ A,
                                                    int Np, int kb) {
  extern __shared__ float t[];  // [128][128]
  int tid = threadIdx.x;
  int r0 = kb * 128;
  for (int c = 0; c < 128; ++c)
    t[tid * 128 + c] = A[(size_t)(r0 + tid) * Np + r0 + c];
  __syncthreads();
  for (int j = 0; j < 128; ++j) {
    if (tid == j) t[j * 128 + j] = sqrtf(t[j * 128 + j]);
    __syncthreads();
    float d = t[j * 128 + j];
    if (tid > j) t[tid * 128 + j] /= d;
    __syncthreads();
    if (tid > j) {
      float l = t[tid * 128 + j];
      for (int c = j + 1; c <= tid; ++c) t[tid * 128 + c] -= l * t[c * 128 + j];
    }
    __syncthreads();
  }
  for (int c = 0; c < 128; ++c)
    A[(size_t)(r0 + tid) * Np + r0 + c] = t[tid * 128 + c];
}

// ---------------------------------------------------------------------------
// TRSM: panel rows r below the diagonal block solve  X * Lkk^T = A_panel.
// One thread per row; row lives in LDS column-major (conflict-free), Lkk rows
// are wave-uniform global loads (broadcast, L2-resident).
// ---------------------------------------------------------------------------
__global__ __launch_bounds__(128) void trsm_kernel(float* __restrict__ A,
                                                   int Np, int kb) {
  extern __shared__ float sX[];  // sX[j*128 + tid]
  int tid = threadIdx.x;
  int k0 = kb * 128;
  int r = k0 + 128 + blockIdx.x * 128 + tid;
  const float* L = A + (size_t)k0 * Np + k0;
  for (int j = 0; j < 128; ++j) sX[j * 128 + tid] = A[(size_t)r * Np + k0 + j];
  for (int j = 0; j < 128; ++j) {
    float v = sX[j * 128 + tid];
    for (int t = 0; t < j; ++t)
      v -= L[(size_t)j * Np + t] * sX[t * 128 + tid];
    sX[j * 128 + tid] = v / L[(size_t)j * Np + j];
  }
  for (int j = 0; j < 128; ++j) A[(size_t)r * Np + k0 + j] = sX[j * 128 + tid];
}

// ---------------------------------------------------------------------------
// SYRK/GEMM trailing update via fp32 WMMA: A[bi,bj] -= P_i * P_j^T  (K = 128).
// 256 threads = 8 waves (2x4), each wave a 64x32 quadrant (4x2 WMMA tiles).
// K staged through LDS in chunks of 32 (<= 34 KB LDS).
// ---------------------------------------------------------------------------
__global__ __launch_bounds__(256) void syrk_kernel(float* __restrict__ A,
                                                   int Np, int kb) {
  int bi = kb + 1 + blockIdx.y;
  int bj = kb + 1 + blockIdx.x;
  if (bi < bj) return;  // lower triangle only
  const int KS = 32, PIT = 33;
  extern __shared__ float smem[];
  float* sPi = smem;              // [128][KS] rows of block-row bi
  float* sPj = smem + 128 * PIT;  // [128][KS] rows of block-row bj

  int tid = threadIdx.x;
  int lane = tid & 31;
  int wave = tid >> 5;
  int lhalf = lane >> 4;
  int l16 = lane & 15;
  int waveM = wave >> 2;  // 0..1 -> 64 rows each
  int waveN = wave & 3;   // 0..3 -> 32 cols each
  int k0 = kb * 128;

  const v8f vzero = {0.f, 0.f, 0.f, 0.f, 0.f, 0.f, 0.f, 0.f};
  v8f acc[4][2];
#pragma unroll
  for (int i = 0; i < 4; ++i)
#pragma unroll
    for (int j = 0; j < 2; ++j) acc[i][j] = vzero;

  for (int kc = 0; kc < 128; kc += KS) {
    __syncthreads();
    for (int i = tid; i < 128 * KS; i += 256) {
      int m = i >> 5;  // /KS
      int k = i & 31;  // %KS
      sPi[m * PIT + k] = A[(size_t)(bi * 128 + m) * Np + k0 + kc + k];
      sPj[m * PIT + k] = A[(size_t)(bj * 128 + m) * Np + k0 + kc + k];
    }
    __syncthreads();
#pragma unroll 2
    for (int ks = 0; ks < KS; ks += 4) {
      int kv = ks + lhalf * 2;
      v2f av[4], bv[2];
#pragma unroll
      for (int tm = 0; tm < 4; ++tm) {
        int mm = waveM * 64 + tm * 16 + l16;
        av[tm].x = sPi[mm * PIT + kv];
        av[tm].y = sPi[mm * PIT + kv + 1];
      }
#pragma unroll
      for (int tn = 0; tn < 2; ++tn) {
        int nn = waveN * 32 + tn * 16 + l16;
        bv[tn].x = sPj[nn * PIT + kv];
        bv[tn].y = sPj[nn * PIT + kv + 1];
      }
#pragma unroll
      for (int tm = 0; tm < 4; ++tm)
#pragma unroll
        for (int tn = 0; tn < 2; ++tn)
          acc[tm][tn] = wmma_f32(av[tm], bv[tn], acc[tm][tn]);
    }
  }

#pragma unroll
  for (int tm = 0; tm < 4; ++tm) {
#pragma unroll
    for (int tn = 0; tn < 2; ++tn) {
      int nn = bj * 128 + waveN * 32 + tn * 16 + l16;
#pragma unroll
      for (int r = 0; r < 8; ++r) {
        int mm = bi * 128 + waveM * 64 + tm * 16 + r + 8 * lhalf;
        A[(size_t)mm * Np + nn] -= acc[tm][tn][r];
      }
    }
  }
}

// ---------------------------------------------------------------------------
// Forward solve on diagonal block: L_kk z = y (64 threads = 64 RHS columns)
// ---------------------------------------------------------------------------
__global__ __launch_bounds__(64) void fwd_diag_kernel(
    const float* __restrict__ A, float* __restrict__ Y, int Np, int Bp,
    int kb) {
  extern __shared__ float sY[];  // [128][64]
  int tid = threadIdx.x;
  int k0 = kb * 128;
  const float* L = A + (size_t)k0 * Np + k0;
  for (int rl = 0; rl < 128; ++rl)
    sY[rl * 64 + tid] = Y[(size_t)(k0 + rl) * Bp + tid];
  for (int rl = 0; rl < 128; ++rl) {
    float v = sY[rl * 64 + tid];
    for (int t = 0; t < rl; ++t)
      v -= L[(size_t)rl * Np + t] * sY[t * 64 + tid];
    sY[rl * 64 + tid] = v / L[(size_t)rl * Np + rl];
  }
  for (int rl = 0; rl < 128; ++rl)
    Y[(size_t)(k0 + rl) * Bp + tid] = sY[rl * 64 + tid];
}

// Y[rows below kb] -= L[rows, kb-block] @ Z[kb-block]
__global__ void fwd_update_kernel(const float* __restrict__ A,
                                  float* __restrict__ Y, int Np, int Bp,
                                  int kb) {
  extern __shared__ float sZ[];  // [128][64]
  int b = threadIdx.x;           // 64
  int rl = threadIdx.y;          // 8
  int k0 = kb * 128;
  int tid = rl * 64 + b;
  for (int i = tid; i < 128 * 64; i += 512)
    sZ[i] = Y[(size_t)(k0 + (i >> 6)) * Bp + (i & 63)];
  __syncthreads();
  int r = k0 + 128 + blockIdx.x * 8 + rl;
  if (r < Np) {
    const float* arow = A + (size_t)r * Np + k0;
    float acc = 0.0f;
    for (int t = 0; t < 128; ++t) acc += arow[t] * sZ[t * 64 + b];
    Y[(size_t)r * Bp + b] -= acc;
  }
}

// Backward solve on diagonal block: L_kk^T w = z
__global__ __launch_bounds__(64) void bwd_diag_kernel(
    const float* __restrict__ A, float* __restrict__ Y, int Np, int Bp,
    int kb) {
  extern __shared__ float sY[];  // [128][64]
  int tid = threadIdx.x;
  int k0 = kb * 128;
  const float* L = A + (size_t)k0 * Np + k0;
  for (int rl = 0; rl < 128; ++rl)
    sY[rl * 64 + tid] = Y[(size_t)(k0 + rl) * Bp + tid];
  for (int rl = 127; rl >= 0; --rl) {
    float v = sY[rl * 64 + tid];
    for (int t = rl + 1; t < 128; ++t)
      v -= L[(size_t)t * Np + rl] * sY[t * 64 + tid];
    sY[rl * 64 + tid] = v / L[(size_t)rl * Np + rl];
  }
  for (int rl = 0; rl < 128; ++rl)
    Y[(size_t)(k0 + rl) * Bp + tid] = sY[rl * 64 + tid];
}

// Y[rows above kb] -= L[kb-block, rows]^T @ W[kb-block]
__global__ void bwd_update_kernel(const float* __restrict__ A,
                                  float* __restrict__ Y, int Np, int Bp,
                                  int kb) {
  extern __shared__ float sW[];  // [128][64]
  int b = threadIdx.x;
  int rl = threadIdx.y;
  int k0 = kb * 128;
  int tid = rl * 64 + b;
  for (int i = tid; i < 128 * 64; i += 512)
    sW[i] = Y[(size_t)(k0 + (i >> 6)) * Bp + (i & 63)];
  __syncthreads();
  int r = blockIdx.x * 8 + rl;
  if (r < k0) {
    float acc = 0.0f;
    for (int t = 0; t < 128; ++t)
      acc += A[(size_t)(k0 + t) * Np + r] * sW[t * 64 + b];
    Y[(size_t)r * Bp + b] -= acc;
  }
}

// out[b, i] = sqrt(s[idx[i]]) * Y[i, b]
__global__ void write_out_kernel(const float* __restrict__ Y,
                                 const float* __restrict__ s,
                                 const int* __restrict__ idx,
                                 float* __restrict__ out, int NI, int Bn,
                                 int Bp) {
  int t = blockIdx.x * blockDim.x + threadIdx.x;
  if (t >= Bn * NI) return;
  int b = t / NI;
  int i = t - b * NI;
  out[t] = sqrtf(s[idx[i]]) * Y[(size_t)i * Bp + b];
}

// ---------------------------------------------------------------------------
extern "C" void kernel_launch(void* const* d_in, const int* in_sizes, int n_in,
                              void* d_out, int out_size, void* d_ws,
                              size_t ws_size, hipStream_t stream) {
  const float* X = (const float*)d_in[0];
  const float* d0 = (const float*)d_in[1];
  const float* w = (const float*)d_in[2];
  const float* s = (const float*)d_in[3];
  const int* idx = (const int*)d_in[4];
  float* out = (float*)d_out;

  const int E = in_sizes[2];
  const int V = in_sizes[3];
  const int NI = in_sizes[4];
  const int Bn = in_sizes[0] / V;

  const int Np = ((NI + 127) / 128) * 128;  // 3968
  const int nb = Np / 128;                  // 31
  const int Bp = 64;

  float* A = (float*)d_ws;            // Np*Np fp32 SPD matrix
  float* Y = A + (size_t)Np * Np;     // Np*Bp rhs/solution

  // rhs in SPD basis (also zero-fills padding)
  {
    int tot = Np * Bp;
    build_rhs_kernel<<<(tot + 255) / 256, 256, 0, stream>>>(X, s, idx, Y, V,
                                                            NI, Bn, Bp, Np);
  }
  // A = I + S^{1/2} d0i^T diag(w) d0i S^{1/2}  (fp32 WMMA GEMM)
  {
    dim3 grid(Np / 64, Np / 64);
    size_t sm = 2 * 32 * 65 * sizeof(float);
    gemm_spd_kernel<<<grid, 128, sm, stream>>>(d0, w, s, idx, A, E, V, NI, Np);
  }
  // Blocked right-looking Cholesky, WMMA trailing updates
  for (int kb = 0; kb < nb; ++kb) {
    potrf_kernel<<<1, 128, 128 * 128 * sizeof(float), stream>>>(A, Np, kb);
    int trail = nb - kb - 1;
    if (trail > 0) {
      trsm_kernel<<<trail, 128, 128 * 128 * sizeof(float), stream>>>(A, Np,
                                                                     kb);
      dim3 g(trail, trail);
      syrk_kernel<<<g, 256, 2 * 128 * 33 * sizeof(float), stream>>>(A, Np, kb);
    }
  }
  // Forward substitution L z = y
  for (int kb = 0; kb < nb; ++kb) {
    fwd_diag_kernel<<<1, 64, 128 * 64 * sizeof(float), stream>>>(A, Y, Np, Bp,
                                                                 kb);
    int rows = Np - (kb + 1) * 128;
    if (rows > 0)
      fwd_update_kernel<<<(rows + 7) / 8, dim3(64, 8),
                          128 * 64 * sizeof(float), stream>>>(A, Y, Np, Bp,
                                                              kb);
  }
  // Backward substitution L^T w = z
  for (int kb = nb - 1; kb >= 0; --kb) {
    bwd_diag_kernel<<<1, 64, 128 * 64 * sizeof(float), stream>>>(A, Y, Np, Bp,
                                                                 kb);
    if (kb > 0) {
      int rows = kb * 128;
      bwd_update_kernel<<<(rows + 7) / 8, dim3(64, 8),
                          128 * 64 * sizeof(float), stream>>>(A, Y, Np, Bp,
                                                              kb);
    }
  }
  // Map back: x = S^{1/2} w, transpose to [B, NI]
  {
    int tot = Bn * NI;
    write_out_kernel<<<(tot + 255) / 256, 256, 0, stream>>>(Y, s, idx, out, NI,
                                                            Bn, Bp);
  }
}